// IDWT1D_63969242906644
// MI455X (gfx1250) — compile-verified
//
#include <hip/hip_runtime.h>
#include <stdint.h>

typedef float        v4f   __attribute__((ext_vector_type(4)));
typedef unsigned int u32x4 __attribute__((ext_vector_type(4)));
typedef int          i32x4 __attribute__((ext_vector_type(4)));
typedef int          i32x8 __attribute__((ext_vector_type(8)));

// Fixed by the reference: x is [32*64 rows, 16384]; approx/detail are 8192 each.
#define MFULL  16384
#define NHALF  (MFULL / 2)          // 8192
#define TILE_M 2048                 // m-values (half-band samples) per block
#define BLOCK  256                  // 8 wave32 waves
#define MPT    (TILE_M / BLOCK)     // 8 m per thread -> 16 outputs
#define NWAVES (BLOCK / 32)         // 8
#define CHUNK  (TILE_M / NWAVES)    // 256 floats (1KB) per wave-issued TDM DMA
#define TPR    (NHALF / TILE_M)     // 4 tiles per row (power of two)

static __device__ __forceinline__ int wrapN(int v) {
  if (v < 0)      v += NHALF;
  if (v >= NHALF) v -= NHALF;
  return v;
}

// Stage `len` contiguous floats from global into LDS using the Tensor Data
// Mover (1-D tile descriptor, D# built per CDNA5 ISA).
// Executed by every wave with wave-uniform operands: TDM ignores EXEC, so an
// unconditional per-wave issue is the only divergence-safe pattern.
static __device__ __forceinline__ void stage_to_lds(const float* gsrc, float* ldst,
                                                    int len, int lane) {
#if __has_builtin(__builtin_amdgcn_tensor_load_to_lds)
  (void)lane;
  uint64_t ga = (uint64_t)(uintptr_t)gsrc;          // 57-bit global byte address
  uint32_t la = (uint32_t)(uintptr_t)ldst;          // low 32 bits = LDS byte offset
  u32x4 g0;
  g0.x = 1u;                                        // count=1 (valid), user mode
  g0.y = la;                                        // lds_addr
  g0.z = (uint32_t)ga;                              // global_addr[31:0]
  g0.w = (uint32_t)((ga >> 32) & 0x1FFFFFFull)      // global_addr[56:32]
       | (2u << 30);                                // type=2 ("image")
  i32x8 g1;
  g1[0] = 2 << 16;                                  // wg_mask=0, data_size=2 (4B)
  g1[1] = (len & 0xFFFF) << 16;                     // tensor_dim0[15:0]
  g1[2] = ((len >> 16) & 0xFFFF) | (1 << 16);       // tensor_dim0[31:16], tensor_dim1=1
  g1[3] = (len & 0xFFFF) << 16;                     // tile_dim0 = len
  g1[4] = 0;                                        // tile_dim1/2 = 0 (unused -> 1D)
  g1[5] = len;                                      // tensor_dim0_stride (Y never >0)
  g1[6] = 0;
  g1[7] = 0;
  i32x4 z4 = {0, 0, 0, 0};
#if __clang_major__ >= 23
  i32x8 z8 = {0, 0, 0, 0, 0, 0, 0, 0};
  __builtin_amdgcn_tensor_load_to_lds(g0, g1, z4, z4, z8, 0);
#else
  __builtin_amdgcn_tensor_load_to_lds(g0, g1, z4, z4, 0);
#endif
#else
  for (int i = lane; i < len; i += 32) ldst[i] = gsrc[i];
#endif
}

__global__ __launch_bounds__(BLOCK) void idwt1d_kernel(
    const float* __restrict__ x, const float* __restrict__ lp,
    const float* __restrict__ hp, const int* __restrict__ zlp,
    const int* __restrict__ zhp, float* __restrict__ out) {
  __shared__ float ldsA[TILE_M];
  __shared__ float ldsD[TILE_M];

  const int tid  = threadIdx.x;
  const int tile = blockIdx.x;
  const int row  = tile >> 2;                 // TPR == 4
  const int t0   = (tile & (TPR - 1)) * TILE_M;

  const float* aRow = x + (size_t)row * MFULL;   // approximation half
  const float* dRow = aRow + NHALF;              // detail half
  float*       oRow = out + (size_t)row * MFULL;

  // ---- TDM stage: every wave DMAs its own 1KB slice of A and of D ----
  {
    const int w    = tid >> 5;                   // wave id (wave32)
    const int lane = tid & 31;
    const int off  = w * CHUNK;
    stage_to_lds(aRow + t0 + off, ldsA + off, CHUNK, lane);
    stage_to_lds(dRow + t0 + off, ldsD + off, CHUNK, lane);
  }
#if __has_builtin(__builtin_amdgcn_s_wait_tensorcnt)
  __builtin_amdgcn_s_wait_tensorcnt(0);
#endif
  __syncthreads();

  float lpv[4], hpv[4];
#pragma unroll
  for (int i = 0; i < 4; ++i) { lpv[i] = lp[i]; hpv[i] = hp[i]; }
  const int zl = zlp[0];   // shifts the hp (detail) branch
  const int zh = zhp[0];   // shifts the lp (approx) branch

  const int l0 = tid * MPT;      // local m within tile
  const int m0 = t0 + l0;        // global m within row

  if (zh == 1 && zl == 1) {
    // Fast path (the harness's parameters): fully vectorized.
    v4f aLo = *(const v4f*)&ldsA[l0];
    v4f aHi = *(const v4f*)&ldsA[l0 + 4];
    v4f dLo = *(const v4f*)&ldsD[l0];
    v4f dHi = *(const v4f*)&ldsD[l0 + 4];
    float a8, d8;
    if (l0 + MPT < TILE_M) { a8 = ldsA[l0 + MPT]; d8 = ldsD[l0 + MPT]; }
    else { const int g = wrapN(m0 + MPT); a8 = aRow[g]; d8 = dRow[g]; }
    const float dm1 = (l0 > 0) ? ldsD[l0 - 1] : dRow[wrapN(m0 - 1)];

    const float av[MPT + 1] = {aLo.x, aLo.y, aLo.z, aLo.w,
                               aHi.x, aHi.y, aHi.z, aHi.w, a8};
    const float dv[MPT + 2] = {dm1, dLo.x, dLo.y, dLo.z, dLo.w,
                               dHi.x, dHi.y, dHi.z, dHi.w, d8};

    v4f o[MPT / 2];
#pragma unroll
    for (int j = 0; j < MPT; ++j) {
      // out[2m]   = lp2*a[m] + lp0*a[m+1] + hp3*d[m-1] + hp1*d[m]
      // out[2m+1] = lp3*a[m] + lp1*a[m+1] + hp2*d[m]   + hp0*d[m+1]
      const float ev = lpv[2] * av[j] + lpv[0] * av[j + 1] +
                       hpv[3] * dv[j] + hpv[1] * dv[j + 1];
      const float ov = lpv[3] * av[j] + lpv[1] * av[j + 1] +
                       hpv[2] * dv[j + 1] + hpv[0] * dv[j + 2];
      o[j >> 1][(j & 1) * 2]     = ev;
      o[j >> 1][(j & 1) * 2 + 1] = ov;
    }
    float* op = oRow + 2 * m0;     // 64B-aligned
#pragma unroll
    for (int k = 0; k < MPT / 2; ++k)
      __builtin_nontemporal_store(o[k], (v4f*)(op + 4 * k));
  } else {
    // General zero_lp / zero_hp path (cold; scalar, still LDS-backed).
    const int ph = zh & 1, qh = (zh >> 1) & 1;
    const int pl = zl & 1, ql = (zl >> 1) & 1;
    const float cA0a = lpv[3 - ph], cA0b = lpv[1 - ph];
    const float cA1a = lpv[2 + ph], cA1b = lpv[ph];
    const float cD0a = hpv[2 + pl], cD0b = hpv[pl];
    const float cD1a = hpv[3 - pl], cD1b = hpv[1 - pl];
    const int e0 = -qh, e1 = 1 - ph - qh;
    const int f0 = -pl - ql, f1 = -ql;
#pragma unroll 1
    for (int j = 0; j < MPT; ++j) {
      const int l = l0 + j, m = m0 + j;
      auto FA = [&](int off) -> float {
        const int ll = l + off;
        return (ll >= 0 && ll < TILE_M) ? ldsA[ll] : aRow[wrapN(m + off)];
      };
      auto FD = [&](int off) -> float {
        const int ll = l + off;
        return (ll >= 0 && ll < TILE_M) ? ldsD[ll] : dRow[wrapN(m + off)];
      };
      const float ev = cA0a * FA(e0) + cA0b * FA(e0 + 1) +
                       cD0a * FD(f0) + cD0b * FD(f0 + 1);
      const float ov = cA1a * FA(e1) + cA1b * FA(e1 + 1) +
                       cD1a * FD(f1) + cD1b * FD(f1 + 1);
      oRow[2 * m]     = ev;
      oRow[2 * m + 1] = ov;
    }
  }
}

extern "C" void kernel_launch(void* const* d_in, const int* in_sizes, int n_in,
                              void* d_out, int out_size, void* d_ws, size_t ws_size,
                              hipStream_t stream) {
  (void)n_in; (void)out_size; (void)d_ws; (void)ws_size;
  const float* x   = (const float*)d_in[0];
  const float* lp  = (const float*)d_in[1];
  const float* hp  = (const float*)d_in[2];
  const int*   zlp = (const int*)d_in[3];
  const int*   zhp = (const int*)d_in[4];
  float*       out = (float*)d_out;

  const int rows = in_sizes[0] / MFULL;    // 32*64 = 2048
  const int grid = rows * TPR;             // 8192 blocks

  idwt1d_kernel<<<grid, BLOCK, 0, stream>>>(x, lp, hp, zlp, zhp, out);
}